// GraphAttentionLayer_12352325943711
// MI455X (gfx1250) — compile-verified
//
#include <hip/hip_runtime.h>

#define N_NODES 3072
#define IN_F    512
#define N_HEADS 8
#define N_HID   64
#define OUT_F   512   // N_HEADS * N_HID
#define SLOPE   0.2f
#define SPLITJ  4
#define JTILES  (N_NODES / 16)            // 192
#define JT_PER_CHUNK (JTILES / SPLITJ)    // 48

typedef __attribute__((ext_vector_type(2))) float v2f;
typedef __attribute__((ext_vector_type(8))) float v8f;

static __device__ __forceinline__ v8f wmma_f32(v2f a, v2f b, v8f c) {
  // D = A(16x4) * B(4x16) + C, exact fp32 (V_WMMA_F32_16X16X4_F32)
  return __builtin_amdgcn_wmma_f32_16x16x4_f32(false, a, false, b, (short)0, c, false, false);
}

// ---------------------------------------------------------------------------
// Kernel 1: g = h @ W^T   (3072x512 = 3072x512 * 512x512)
// Block: 256 thr (8 waves). blockIdx.x = 16-row tile, blockIdx.y = 128-col group,
// each wave owns a 16x16 output tile, K loop in steps of 4.
// ---------------------------------------------------------------------------
__global__ __launch_bounds__(256) void gat_gemm_g(
    const float* __restrict__ h, const float* __restrict__ W, float* __restrict__ g) {
  const int lane = threadIdx.x & 31;
  const int wv   = threadIdx.x >> 5;
  const int m16  = lane & 15;
  const int khalf = (lane < 16) ? 0 : 2;           // K offset per ISA A/B layout
  const int row0 = blockIdx.x * 16;
  const int col0 = blockIdx.y * 128 + wv * 16;

  const float* hrow = h + (size_t)(row0 + m16) * IN_F + khalf;  // A[M][K]
  const float* wrow = W + (size_t)(col0 + m16) * IN_F + khalf;  // B[K][N] = W[col0+N][k+K]

  v8f acc = {};
  #pragma unroll 4
  for (int k = 0; k < IN_F; k += 4) {
    v2f a = *(const v2f*)(hrow + k);   // {A[M][k+khalf], A[M][k+khalf+1]}  8B aligned
    v2f b = *(const v2f*)(wrow + k);
    acc = wmma_f32(a, b, acc);
  }
  // C/D layout: VGPR r -> M = r + 8*(lane>=16), N = lane&15
  const int mbase = row0 + ((lane < 16) ? 0 : 8);
  #pragma unroll
  for (int r = 0; r < 8; ++r)
    g[(size_t)(mbase + r) * OUT_F + col0 + m16] = acc[r];
}

// ---------------------------------------------------------------------------
// Kernel 2: s_src[j,h] = g[j,h,:].a_src ; s_dst[j,h] = g[j,h,:].a_dst
// One block per node j, wave w = head w, 64-wide dot via wave32 shuffle reduce.
// ---------------------------------------------------------------------------
__global__ __launch_bounds__(256) void gat_scores(
    const float* __restrict__ g, const float* __restrict__ aw,
    float* __restrict__ ssrc, float* __restrict__ sdst) {
  const int j = blockIdx.x;
  const int lane = threadIdx.x & 31;
  const int hh = threadIdx.x >> 5;
  const float* grow = g + (size_t)j * OUT_F + hh * N_HID;
  const float x0 = grow[lane], x1 = grow[lane + 32];
  float ps = x0 * aw[lane]      + x1 * aw[lane + 32];
  float pd = x0 * aw[64 + lane] + x1 * aw[96 + lane];
  #pragma unroll
  for (int off = 16; off > 0; off >>= 1) {
    ps += __shfl_xor(ps, off, 32);
    pd += __shfl_xor(pd, off, 32);
  }
  if (lane == 0) { ssrc[(size_t)j * N_HEADS + hh] = ps; sdst[(size_t)j * N_HEADS + hh] = pd; }
}

// ---------------------------------------------------------------------------
// Kernel 3: masked-softmax row statistics: m[i,h] = max_j e, invd[i,h] = 1/sum_j exp(e-m)
// One block per row i; online (max,sum) per thread, then wave + LDS combine.
// ---------------------------------------------------------------------------
__global__ __launch_bounds__(256) void gat_rowstats(
    const int* __restrict__ adj, const float* __restrict__ ssrc,
    const float* __restrict__ sdst, float* __restrict__ mrow, float* __restrict__ invd) {
  const int i = blockIdx.x;
  const int t = threadIdx.x;
  float sd[8], mh[8], sh[8];
  #pragma unroll
  for (int hh = 0; hh < 8; ++hh) {
    sd[hh] = sdst[(size_t)i * N_HEADS + hh];
    mh[hh] = -__builtin_inff();
    sh[hh] = 0.f;
  }
  const int* arow = adj + (size_t)i * N_NODES;
  for (int j = t; j < N_NODES; j += 256) {
    if (arow[j]) {
      const float* sr = ssrc + (size_t)j * N_HEADS;
      #pragma unroll
      for (int hh = 0; hh < 8; ++hh) {
        float x = sd[hh] + sr[hh];
        float e = (x >= 0.f) ? x : SLOPE * x;
        if (e > mh[hh]) { sh[hh] = sh[hh] * __expf(mh[hh] - e) + 1.f; mh[hh] = e; }
        else            { sh[hh] += __expf(e - mh[hh]); }
      }
    }
  }
  // wave32 combine of (m,s) pairs
  #pragma unroll
  for (int off = 16; off > 0; off >>= 1) {
    #pragma unroll
    for (int hh = 0; hh < 8; ++hh) {
      float om = __shfl_xor(mh[hh], off, 32);
      float os = __shfl_xor(sh[hh], off, 32);
      float M  = fmaxf(mh[hh], om);
      float s1 = (sh[hh] == 0.f) ? 0.f : sh[hh] * __expf(mh[hh] - M);
      float s2 = (os     == 0.f) ? 0.f : os     * __expf(om     - M);
      mh[hh] = M; sh[hh] = s1 + s2;
    }
  }
  __shared__ float lm[8][8], ls[8][8];   // [wave][head]
  const int wv = t >> 5, lane = t & 31;
  if (lane == 0) {
    #pragma unroll
    for (int hh = 0; hh < 8; ++hh) { lm[wv][hh] = mh[hh]; ls[wv][hh] = sh[hh]; }
  }
  __syncthreads();
  if (t < 8) {   // t = head
    float M = -__builtin_inff(), S = 0.f;
    #pragma unroll
    for (int w2 = 0; w2 < 8; ++w2) {
      float m2 = lm[w2][t], s2 = ls[w2][t];
      float Mn = fmaxf(M, m2);
      float a = (S  == 0.f) ? 0.f : S  * __expf(M  - Mn);
      float b = (s2 == 0.f) ? 0.f : s2 * __expf(m2 - Mn);
      M = Mn; S = a + b;
    }
    mrow[(size_t)i * N_HEADS + t] = M;
    invd[(size_t)i * N_HEADS + t] = 1.f / S;
  }
}

// ---------------------------------------------------------------------------
// Kernel 4: out_partial[i,h,:] += sum over j-chunk of p[i,j,h]*g[j,h,:] via f32 WMMA.
// Block: 16 i-rows x all 8 heads (wave = head), blockIdx.y = j-chunk.
// p built on the fly from rank-1 scores + adj tile staged in LDS.
// ---------------------------------------------------------------------------
__global__ __launch_bounds__(256) void gat_aggregate(
    const int* __restrict__ adj, const float* __restrict__ g,
    const float* __restrict__ ssrc, const float* __restrict__ sdst,
    const float* __restrict__ mrow, const float* __restrict__ invd,
    float* __restrict__ part) {
  __shared__ int adjtile[256];
  const int lane = threadIdx.x & 31;
  const int hh   = threadIdx.x >> 5;
  const int m16  = lane & 15;
  const int hi   = (lane >= 16) ? 1 : 0;
  const int i0   = blockIdx.x * 16;
  const int jt0  = blockIdx.y * JT_PER_CHUNK;

  const float sdv = sdst[(size_t)(i0 + m16) * N_HEADS + hh];
  const float mv  = mrow[(size_t)(i0 + m16) * N_HEADS + hh];
  const float idv = invd[(size_t)(i0 + m16) * N_HEADS + hh];

  const int lr = threadIdx.x >> 4;  // cooperative adj-tile load coords
  const int lc = threadIdx.x & 15;

  v8f acc0 = {}, acc1 = {}, acc2 = {}, acc3 = {};

  for (int jt = jt0; jt < jt0 + JT_PER_CHUNK; ++jt) {
    const int j0 = jt * 16;
    __syncthreads();
    adjtile[threadIdx.x] = adj[(size_t)(i0 + lr) * N_NODES + j0 + lc];
    if (jt + 1 < jt0 + JT_PER_CHUNK)   // global_prefetch_b8 for next adj tile
      __builtin_prefetch(&adj[(size_t)(i0 + lr) * N_NODES + j0 + 16 + lc], 0, 1);
    __syncthreads();

    #pragma unroll
    for (int kk = 0; kk < 4; ++kk) {          // four K=4 WMMA steps cover 16 j's
      const int klo = kk * 4 + hi * 2;        // per-lane K base per ISA A layout
      const int j   = j0 + klo;
      const int a0  = adjtile[m16 * 16 + klo];
      const int a1  = adjtile[m16 * 16 + klo + 1];
      const float ss0 = ssrc[(size_t)j * N_HEADS + hh];
      const float ss1 = ssrc[(size_t)(j + 1) * N_HEADS + hh];
      float x0 = sdv + ss0; x0 = (x0 >= 0.f) ? x0 : SLOPE * x0;
      float x1 = sdv + ss1; x1 = (x1 >= 0.f) ? x1 : SLOPE * x1;
      v2f A;
      A.x = a0 ? __expf(x0 - mv) * idv : 0.f;   // normalized attention weight
      A.y = a1 ? __expf(x1 - mv) * idv : 0.f;
      const float* gb = g + (size_t)j * OUT_F + hh * N_HID + m16;  // B[K][N]=g[j0+K][...]
      v2f B0 = { gb[0],  gb[OUT_F] };
      v2f B1 = { gb[16], gb[16 + OUT_F] };
      v2f B2 = { gb[32], gb[32 + OUT_F] };
      v2f B3 = { gb[48], gb[48 + OUT_F] };
      acc0 = wmma_f32(A, B0, acc0);
      acc1 = wmma_f32(A, B1, acc1);
      acc2 = wmma_f32(A, B2, acc2);
      acc3 = wmma_f32(A, B3, acc3);
    }
  }

  float* prow = part + (size_t)blockIdx.y * ((size_t)N_NODES * OUT_F);
  const int mbase = i0 + hi * 8;
  #pragma unroll
  for (int r = 0; r < 8; ++r) {
    float* p = prow + (size_t)(mbase + r) * OUT_F + hh * N_HID + m16;
    p[0]  = acc0[r];
    p[16] = acc1[r];
    p[32] = acc2[r];
    p[48] = acc3[r];
  }
}

// ---------------------------------------------------------------------------
// Kernel 5: deterministic combine of the SPLITJ j-chunk partials.
// ---------------------------------------------------------------------------
__global__ __launch_bounds__(256) void gat_reduce(
    const float* __restrict__ part, float* __restrict__ out) {
  const size_t G = (size_t)N_NODES * OUT_F;
  const size_t idx = (size_t)blockIdx.x * 256 + threadIdx.x;
  float s = part[idx] + part[G + idx] + part[2 * G + idx] + part[3 * G + idx];
  out[idx] = s;
}

// ---------------------------------------------------------------------------
extern "C" void kernel_launch(void* const* d_in, const int* in_sizes, int n_in,
                              void* d_out, int out_size, void* d_ws, size_t ws_size,
                              hipStream_t stream) {
  const float* h   = (const float*)d_in[0];
  const int*   adj = (const int*)d_in[1];
  const float* W   = (const float*)d_in[2];
  const float* aw  = (const float*)d_in[3];
  float* out = (float*)d_out;

  float* ws = (float*)d_ws;
  const size_t G = (size_t)N_NODES * OUT_F;       // 1,572,864 floats
  const size_t S = (size_t)N_NODES * N_HEADS;     //    24,576 floats
  float* g    = ws;                // G
  float* ssrc = g    + G;          // S
  float* sdst = ssrc + S;          // S
  float* mrow = sdst + S;          // S
  float* invd = mrow + S;          // S
  float* part = invd + S;          // SPLITJ * G  (total ws ~= 31.9 MB)

  gat_gemm_g   <<<dim3(N_NODES / 16, OUT_F / 128), 256, 0, stream>>>(h, W, g);
  gat_scores   <<<N_NODES, 256, 0, stream>>>(g, aw, ssrc, sdst);
  gat_rowstats <<<N_NODES, 256, 0, stream>>>(adj, ssrc, sdst, mrow, invd);
  gat_aggregate<<<dim3(N_NODES / 16, SPLITJ), 256, 0, stream>>>(adj, g, ssrc, sdst, mrow, invd, part);
  gat_reduce   <<<(int)(G / 256), 256, 0, stream>>>(part, out);
}